// UnpillarNetwork_25881472926248
// MI455X (gfx1250) — compile-verified
//
#include <hip/hip_runtime.h>

// CDNA5 / gfx1250, wave32. V_WMMA_F32_16X16X4_F32 for the 128->32 layer.
// Grid table kept L2-resident; all stream-once traffic is non-temporal.

typedef __attribute__((ext_vector_type(2))) float v2f;
typedef __attribute__((ext_vector_type(4))) float v4f;
typedef __attribute__((ext_vector_type(8))) float v8f;

constexpr int NPTS   = 1000000;
constexpr int PLANE  = 512 * 512;   // elements per grid feature plane
constexpr int KDIM   = 128;         // concat feature dim
constexpr int RS     = 132;         // padded row stride (A and B tiles): (4r+k) banking
constexpr int HS     = 34;          // padded h-tile row stride
constexpr int WAVES  = 4;           // waves per block
constexpr int TILE_M = 16;          // points per wave

constexpr int BW_SZ  = 32 * RS;               // Yw staged: 32 rows x RS
constexpr int A_SZ   = TILE_M * RS;           // per-wave A tile (h tile aliases it)
constexpr int LDS_FLOATS = BW_SZ + WAVES * A_SZ;   // 12672 floats = 49.5 KB

__global__ __launch_bounds__(WAVES * 32) void unpillar_kernel(
    const float* __restrict__ grid,   // [64][512*512]
    const float* __restrict__ pc,     // [N][64]
    const int*   __restrict__ idx,    // [N]
    const float* __restrict__ Yw,     // [32][128]
    const float* __restrict__ Yb,     // [32]
    const float* __restrict__ Zw,     // [3][32]
    const float* __restrict__ Zb,     // [3]
    float*       __restrict__ out)    // [N][3]
{
    __shared__ float S[LDS_FLOATS];

    const int tid  = threadIdx.x;
    const int wave = tid >> 5;
    const int lane = tid & 31;
    const int half = lane >> 4;       // 0: lanes 0-15, 1: lanes 16-31
    const int rr   = lane & 15;
    const int pb   = (blockIdx.x * WAVES + wave) * TILE_M;   // tile base point

    float* Bw = S;                          // [32][RS]  Yw staged, padded
    float* A  = S + BW_SZ + wave * A_SZ;    // [16][RS]  this wave's A tile
    float* H  = A;                          // [16][HS]  h tile aliases A (A dead by then)

    // ---- Stage Yw -> LDS once per block (coalesced b128, padded stride) ----
    #pragma unroll
    for (int s = 0; s < 8; ++s) {
        const int f   = tid + 128 * s;      // float4 id, 0..1023
        const int row = f >> 5;             // 32 float4 per Yw row
        const int q   = f & 31;
        const v4f v = ((const v4f*)Yw)[f];  // temporal: L2-shared across 15625 blocks
        *(v4f*)&Bw[row * RS + 4 * q] = v;   // 16B aligned: 528*row + 16*q
    }

    // ---- Stage A-tile cols 0..63: random gather (the real bottleneck) ------
    // grid_flat[p][c] = grid[c*PLANE + p]; two feature planes per iteration.
    const int gi = __builtin_nontemporal_load(idx + pb + rr);
    #pragma unroll 4
    for (int s = 0; s < 32; ++s) {
        const int c = 2 * s + half;
        A[rr * RS + c] = grid[(long long)c * PLANE + gi];   // temporal: keep table in L2
    }

    // ---- Stage A-tile cols 64..127: point cloud, stream-once -> NT ---------
    const v4f* pc4 = (const v4f*)pc + (long long)pb * 16;   // wave's 4 KB block
    #pragma unroll
    for (int s = 0; s < 8; ++s) {
        const int pos = lane + 32 * s;      // 0..255 float4 chunks
        const int r   = pos >> 4;
        const int q   = pos & 15;
        const v4f v = __builtin_nontemporal_load(pc4 + pos);
        *(v4f*)&A[r * RS + 64 + 4 * q] = v;
    }

    __syncthreads();   // publish Bw (A/h tiles are wave-private, DS is in-order per wave)

    // ---- Layer 1: D(16x32) = A(16x128) x Yw^T via V_WMMA_F32_16X16X4_F32 ---
    // A frag: lane(half,rr) holds A[rr][4s+2*half + {0,1}]
    // B frag: lane(half,rr) holds B[4s+2*half + {0,1}][n], B[k][n] = Yw[n][k]
    v8f c0 = {};   // columns n = 0..15
    v8f c1 = {};   // columns n = 16..31
    const float* Ar    = A  + rr * RS;
    const float* Brow0 = Bw + rr * RS;
    const float* Brow1 = Bw + (rr + 16) * RS;
    #pragma unroll 4
    for (int s = 0; s < 32; ++s) {
        const int k = 4 * s + 2 * half;
        const v2f a  = *(const v2f*)&Ar[k];
        const v2f b0 = *(const v2f*)&Brow0[k];
        const v2f b1 = *(const v2f*)&Brow1[k];
        c0 = __builtin_amdgcn_wmma_f32_16x16x4_f32(false, a, false, b0,
                                                   (short)0, c0, false, false);
        c1 = __builtin_amdgcn_wmma_f32_16x16x4_f32(false, a, false, b1,
                                                   (short)0, c1, false, false);
    }

    // ---- Fold in Y_b, spill h over the dead A tile -------------------------
    const float yb0 = Yb[rr];
    const float yb1 = Yb[rr + 16];
    #pragma unroll
    for (int j = 0; j < 8; ++j) {
        const int m = j + 8 * half;         // C layout: VGPR j -> row j + 8*half
        H[m * HS + rr]      = c0[j] + yb0;
        H[m * HS + rr + 16] = c1[j] + yb1;
    }

    // ---- Layer 2: 48 contiguous outputs per tile; coalesced NT stores ------
    #pragma unroll
    for (int it = 0; it < 2; ++it) {
        const int p = lane + 32 * it;       // flat output id within tile
        if (p < 48) {
            const int r = p / 3;
            const int d = p - 3 * r;
            float acc = Zb[d];
            #pragma unroll 8
            for (int n = 0; n < 32; ++n)    // same-address lanes broadcast in LDS
                acc = fmaf(H[r * HS + n], Zw[d * 32 + n], acc);
            __builtin_nontemporal_store(acc, out + (long long)pb * 3 + p);
        }
    }
}

extern "C" void kernel_launch(void* const* d_in, const int* in_sizes, int n_in,
                              void* d_out, int out_size, void* d_ws, size_t ws_size,
                              hipStream_t stream) {
    const float* grid = (const float*)d_in[0];   // 64*512*512 f32
    const float* pc   = (const float*)d_in[1];   // 1e6*64 f32
    const int*   idx  = (const int*)d_in[2];     // 1e6 int
    const float* Yw   = (const float*)d_in[3];   // 32*128 f32
    const float* Yb   = (const float*)d_in[4];   // 32 f32
    const float* Zw   = (const float*)d_in[5];   // 3*32 f32
    const float* Zb   = (const float*)d_in[6];   // 3 f32
    float* out = (float*)d_out;                  // 1e6*3 f32

    const int nblocks = NPTS / (WAVES * TILE_M); // 15625, exact
    unpillar_kernel<<<nblocks, WAVES * 32, 0, stream>>>(grid, pc, idx, Yw, Yb, Zw, Zb, out);
}